// StateTrackingMemory_41549513621991
// MI455X (gfx1250) — compile-verified
//
#include <hip/hip_runtime.h>
#include <hip/hip_bf16.h>

// Problem constants (match reference)
#define Bc    8
#define Tc    2048
#define Dc    1024
#define Ec    4
#define Kc    64
#define NTAGS 512
#define G3D   (3 * Dc)   // 3072

typedef __attribute__((ext_vector_type(2))) float v2f;
typedef __attribute__((ext_vector_type(8))) float v8f;

// ---------------------------------------------------------------------------
// Kernel 1: zero the whole output (injection is sparse: only 512 rows written)
// ---------------------------------------------------------------------------
__global__ void stm_zero_kernel(float* __restrict__ out, size_t n) {
    size_t i = (size_t)blockIdx.x * blockDim.x + threadIdx.x;
    size_t stride = (size_t)gridDim.x * blockDim.x;
    for (; i < n; i += stride) out[i] = 0.0f;
}

// ---------------------------------------------------------------------------
// Kernel 2: GI = gather(h)[512x1024] @ W_ih^T [1024x3072] + b_ih  via f32 WMMA
// One 16x16 output tile per wave; K=1024 -> 256 x v_wmma_f32_16x16x4_f32.
// A layout (ISA 7.12.2, 32-bit A 16x4): lanes 0-15 -> K=0(v0),K=1(v1);
// lanes 16-31 -> K=2(v0),K=3(v1); M = lane%16. B symmetric with N = lane%16.
// D layout: VGPR g, lanes 0-15: M=g, N=lane; lanes 16-31: M=g+8, N=lane-16.
// ---------------------------------------------------------------------------
__global__ __launch_bounds__(256)
void stm_gi_gemm_kernel(const float* __restrict__ h,
                        const int*   __restrict__ batch_idx,
                        const int*   __restrict__ seq_pos,
                        const float* __restrict__ W_ih,
                        const float* __restrict__ b_ih,
                        float*       __restrict__ GI) {
    const int lane = threadIdx.x & 31;
    const int wave = threadIdx.x >> 5;
    const int tile = blockIdx.x * 8 + wave;       // 6144 tiles total
    const int tr   = tile / (G3D / 16);           // tag-tile row   (0..31)
    const int tc   = tile % (G3D / 16);           // out-tile col   (0..191)
    const int t0   = tr * 16;
    const int j0   = tc * 16;

    const int mn    = lane & 15;                  // A row M / B col N
    const int khalf = (lane >> 4) << 1;           // 0 or 2

    // Gather A row directly from h[batch_idx[t], seq_pos[t], :]
    const int    tagRow = t0 + mn;
    const size_t aBase  = ((size_t)batch_idx[tagRow] * Tc + (size_t)seq_pos[tagRow]) * Dc;
    const float* aPtr   = h + aBase;
    const float* bPtr   = W_ih + (size_t)(j0 + mn) * Dc;   // row j of W_ih = col j of W_ih^T

    v8f c = {};
    #pragma unroll 4
    for (int k0 = 0; k0 < Dc; k0 += 4) {
        v2f a  = *(const v2f*)(aPtr + k0 + khalf);
        v2f bb = *(const v2f*)(bPtr + k0 + khalf);
        // (neg_a, A, neg_b, B, c_mod, C, reuse_a, reuse_b)
        c = __builtin_amdgcn_wmma_f32_16x16x4_f32(false, a, false, bb,
                                                  (short)0, c, false, false);
    }

    const float bias     = b_ih[j0 + mn];
    const int   mBaseOut = (lane >> 4) * 8;
    #pragma unroll
    for (int g = 0; g < 8; ++g) {
        const int mo = g + mBaseOut;
        GI[(size_t)(t0 + mo) * G3D + (j0 + mn)] = c[g] + bias;
    }
}

// ---------------------------------------------------------------------------
// Kernel 3: sequential per-batch chain. One block per batch (8 blocks x 1024).
// Entity states live in LDS; VQ distance reduction via ds_add_f32 atomics.
// ---------------------------------------------------------------------------
__global__ __launch_bounds__(1024)
void stm_chain_kernel(const float* __restrict__ h,
                      const int*   __restrict__ batch_idx,
                      const int*   __restrict__ seq_pos,
                      const int*   __restrict__ token_id,
                      const int*   __restrict__ char_tag_id,
                      const float* __restrict__ W_hh,
                      const float* __restrict__ b_hh,
                      const float* __restrict__ codebook,
                      const float* __restrict__ W_inj,
                      const float* __restrict__ GI,
                      float*       __restrict__ out,
                      float*       __restrict__ partC,
                      int*         __restrict__ partN) {
    __shared__ float s_states[Ec][Dc];
    __shared__ float s_dist[Kc];
    __shared__ int   s_idx;
    __shared__ float s_commit;

    const int b    = blockIdx.x;
    const int t    = threadIdx.x;              // element index 0..1023
    const int ctag = char_tag_id[0];

    #pragma unroll
    for (int e = 0; e < Ec; ++e) s_states[e][t] = 0.0f;

    int   act  = 0;      // replicated scalar, identical in every thread
    float totC = 0.0f;
    int   nU   = 0;
    __syncthreads();

    for (int i = 0; i < NTAGS; ++i) {
        const int bi = batch_idx[i];           // uniform across block
        if (bi != b) continue;
        const int  p       = seq_pos[i];
        const int  tid     = token_id[i];
        const bool is_char = (tid == ctag);
        const int  slot_upd = (act + Ec - 1) % Ec;

        // ---- GRU gates, element t (gi precomputed incl. bias) ----
        const size_t giBase = (size_t)i * G3D;
        const float gi_r = GI[giBase + t];
        const float gi_z = GI[giBase + Dc + t];
        const float gi_n = GI[giBase + 2 * Dc + t];

        const float* cur = s_states[slot_upd];
        float gh_r = b_hh[t];
        float gh_z = b_hh[Dc + t];
        float gh_n = b_hh[2 * Dc + t];
        const float* wr = W_hh + (size_t)t * Dc;
        const float* wz = W_hh + (size_t)(Dc + t) * Dc;
        const float* wn = W_hh + (size_t)(2 * Dc + t) * Dc;
        for (int k = 0; k < Dc; ++k) {
            const float ck = cur[k];           // LDS broadcast read
            gh_r = fmaf(wr[k], ck, gh_r);
            gh_z = fmaf(wz[k], ck, gh_z);
            gh_n = fmaf(wn[k], ck, gh_n);
        }
        const float cur_t = cur[t];
        const float r  = 1.0f / (1.0f + __expf(-(gi_r + gh_r)));
        const float zz = 1.0f / (1.0f + __expf(-(gi_z + gh_z)));
        const float nn = tanhf(gi_n + r * gh_n);
        const float newst = (1.0f - zz) * nn + zz * cur_t;

        const float x_t = h[((size_t)b * Tc + p) * Dc + t];
        const float z_t = is_char ? x_t : newst;

        // ---- VQ: nearest codebook entry (L2) ----
        if (t < Kc) s_dist[t] = 0.0f;
        __syncthreads();
        for (int c = 0; c < Kc; ++c) {
            const float d = z_t - codebook[(size_t)c * Dc + t];
            atomicAdd(&s_dist[c], d * d);      // ds_add_f32
        }
        __syncthreads();
        if (t == 0) {
            int   best = 0;
            float bd   = s_dist[0];
            for (int c = 1; c < Kc; ++c) {
                if (s_dist[c] < bd) { bd = s_dist[c]; best = c; }
            }
            s_idx    = best;
            s_commit = bd * (1.0f / (float)Dc);
        }
        __syncthreads();                        // also orders cur-reads before write
        const int   idx  = s_idx;
        const float zq   = codebook[(size_t)idx * Dc + t];
        const float zqst = z_t + (zq - z_t);    // straight-through fwd value

        // ---- state write ----
        const bool do_upd = is_char || (act > 0);
        const int  slot   = is_char ? (act % Ec) : slot_upd;
        if (do_upd) s_states[slot][t] = zqst;
        __syncthreads();

        // ---- injection matvec, row t ----
        const int act_new = act + (is_char ? 1 : 0);
        if (act_new > 0) {
            const int    slot_inj = (act_new + Ec - 1) % Ec;
            const float* st = s_states[slot_inj];
            const float* wj = W_inj + (size_t)t * Dc;
            float acc = 0.0f;
            for (int k = 0; k < Dc; ++k) acc = fmaf(wj[k], st[k], acc);
            out[((size_t)b * Tc + p) * Dc + t] = acc;
        }

        if (do_upd) { totC += s_commit; nU += 1; }
        act = act_new;
        __syncthreads();
    }

    if (t == 0) { partC[b] = totC; partN[b] = nU; }
}

// ---------------------------------------------------------------------------
// Kernel 4: deterministic reduction of per-batch commit partials
// ---------------------------------------------------------------------------
__global__ void stm_finalize_kernel(const float* __restrict__ partC,
                                    const int*   __restrict__ partN,
                                    float*       __restrict__ out_avg) {
    if (threadIdx.x == 0 && blockIdx.x == 0) {
        float tc = 0.0f;
        int   n  = 0;
        for (int b = 0; b < Bc; ++b) { tc += partC[b]; n += partN[b]; }
        out_avg[0] = tc / (float)(n > 1 ? n : 1);
    }
}

// ---------------------------------------------------------------------------
extern "C" void kernel_launch(void* const* d_in, const int* in_sizes, int n_in,
                              void* d_out, int out_size, void* d_ws, size_t ws_size,
                              hipStream_t stream) {
    const float* h        = (const float*)d_in[0];
    const int*   batchIdx = (const int*)d_in[1];
    const int*   seqPos   = (const int*)d_in[2];
    const int*   tokenId  = (const int*)d_in[3];
    const int*   charTag  = (const int*)d_in[4];
    const float* W_ih     = (const float*)d_in[5];
    const float* W_hh     = (const float*)d_in[6];
    const float* b_ih     = (const float*)d_in[7];
    const float* b_hh     = (const float*)d_in[8];
    const float* codebook = (const float*)d_in[9];
    const float* W_inj    = (const float*)d_in[10];
    float* out = (float*)d_out;

    // workspace layout: GI [512*3072 f32] | partC [8 f32] | partN [8 i32]
    float* GI    = (float*)d_ws;
    float* partC = GI + (size_t)NTAGS * G3D;
    int*   partN = (int*)(partC + Bc);

    // 1) zero the (sparse) injection output + avg slot
    stm_zero_kernel<<<2048, 256, 0, stream>>>(out, (size_t)out_size);

    // 2) bulk GEMM on WMMA: GI = gather(h) @ W_ih^T + b_ih
    //    6144 tiles of 16x16, 8 waves/block -> 768 blocks
    stm_gi_gemm_kernel<<<768, 256, 0, stream>>>(h, batchIdx, seqPos, W_ih, b_ih, GI);

    // 3) sequential per-batch chains (independent -> 8 blocks)
    stm_chain_kernel<<<Bc, Dc, 0, stream>>>(h, batchIdx, seqPos, tokenId, charTag,
                                            W_hh, b_hh, codebook, W_inj, GI,
                                            out, partC, partN);

    // 4) avg_commit (deterministic)
    stm_finalize_kernel<<<1, 64, 0, stream>>>(partC, partN, out + (out_size - 1));
}